// MTA_32134945308884
// MI455X (gfx1250) — compile-verified
//
#include <hip/hip_runtime.h>

// ---- CDNA5 / gfx1250: wave32, WMMA f32 16x16x4 (full-precision matrix path) ----
typedef __attribute__((ext_vector_type(2))) float v2f;
typedef __attribute__((ext_vector_type(4))) float v4f;
typedef __attribute__((ext_vector_type(8))) float v8f;

#define N_   64
#define C_   256
#define C4_  64
#define V_   25
#define T_   128
#define K_   3
#define TK_  5
#define OCH_ 192              // C4*K
#define EPS_ 1e-5f
#define SLOPE_ 0.2f

#define S_ELEMS (N_*C4_*V_*T_)       // 13,107,200 floats (acc/conv intermediate)
#define Y_ELEMS (N_*OCH_*V_*T_)      // 39,321,600 floats (per-node GEMM output)

static __device__ __forceinline__ v8f wmma_f32(v2f a, v2f b, v8f c) {
  return __builtin_amdgcn_wmma_f32_16x16x4_f32(false, a, false, b,
                                               (short)0, c, false, false);
}

// ---------------------------------------------------------------------------
// Phase A: acc = split[br] (+ z[br-1] from d_out), depthwise conv over T + bias
// s layout: [n, c, v, t] contiguous in t.
// ---------------------------------------------------------------------------
__global__ void MTA_tconv_kernel(const float* __restrict__ x,
                                 const float* __restrict__ outbuf,
                                 const float* __restrict__ cw,   // [3, C4, 5]
                                 const float* __restrict__ cb,   // [3, C4]
                                 float* __restrict__ s, int br) {
  int idx = blockIdx.x * blockDim.x + threadIdx.x;
  if (idx >= S_ELEMS) return;
  int t  = idx & (T_ - 1);
  int v  = (idx >> 7) % V_;
  int cn = idx / (V_ * T_);
  int c  = cn & (C4_ - 1);
  int n  = cn >> 6;

  int cprev = (br > 0 ? br - 1 : 0);
  const float* xb = x      + (((size_t)n * C_ + br    * C4_ + c) * V_ + v) * T_;
  const float* zb = outbuf + (((size_t)n * C_ + cprev * C4_ + c) * V_ + v) * T_;

  float r = cb[br * C4_ + c];
#pragma unroll
  for (int j = 0; j < TK_; ++j) {
    int tt = t + j - 2;
    float val = 0.0f;
    if (tt >= 0 && tt < T_) {
      val = xb[tt];
      if (br > 0) val += zb[tt];
    }
    r += cw[(br * C4_ + c) * TK_ + j] * val;
  }
  s[idx] = r;
}

// ---------------------------------------------------------------------------
// Phase B: per-node GEMM.  For node v: Y_v[o=192, m=N*T] = W_v[192x64] x S_v[64, m]
// One wave owns one (v, 16-wide column strip).  The 16 B-operand k-step
// fragments (the s data) are loaded ONCE into 32 VGPRs, then all 12 M-tiles
// of W_v are swept against them -> s is read exactly once per element.
// Fragment layouts per CDNA5 ISA 7.12.2:
//   A 16x4:  m = lane&15; VGPR j holds K = j + 2*(lane>>4)
//   B 4x16:  n = lane&15; VGPR j holds K = j + 2*(lane>>4)
//   D 16x16: VGPR r holds row r (lanes 0-15) / r+8 (lanes 16-31)
// ---------------------------------------------------------------------------
__global__ void MTA_gemm_node_kernel(const float* __restrict__ s,
                                     const float* __restrict__ gw,  // [3, V, 192, 64]
                                     float* __restrict__ y, int br) {
  int lane = threadIdx.x & 31;
  int wg = blockIdx.x * (blockDim.x >> 5) + (threadIdx.x >> 5);
  // waves: v(25) x mt(512)
  int mt = wg % 512;
  int v  = wg / 512;
  if (v >= V_) return;                 // wave-uniform

  int half = lane >> 4;
  int l15  = lane & 15;
  int col  = mt * 16 + l15;            // m = n*T + t (strip never crosses n)
  int n = col >> 7;
  int t = col & (T_ - 1);

  // ---- load all 16 B k-step fragments into registers (one pass over s) ----
  const float* Sv = s + ((size_t)n * C4_ * V_ + v) * T_ + t;   // + c * V_*T_
  v2f bfr[16];
#pragma unroll
  for (int kk = 0; kk < 16; ++kk) {
    int kb = kk * 4 + half * 2;
    bfr[kk].x = Sv[(size_t)(kb    ) * (V_ * T_)];
    bfr[kk].y = Sv[(size_t)(kb + 1) * (V_ * T_)];
  }

  const float* Wbase = gw + ((size_t)(br * V_ + v) * OCH_ + l15) * C4_;
  float* yb = y + (size_t)n * OCH_ * V_ * T_ + (size_t)v * T_ + t;

  // ---- sweep the 12 M-tiles of W_v against the resident B fragments ----
  for (int mo = 0; mo < 12; ++mo) {
    const float* Wrow = Wbase + (size_t)mo * 16 * C4_;
    if (mo < 11) __builtin_prefetch(Wrow + 16 * C4_, 0, 0);   // next M-tile row
    v8f d = {};
#pragma unroll
    for (int kk = 0; kk < 16; ++kk) {
      int kb = kk * 4 + half * 2;
      v2f a = *(const v2f*)(Wrow + kb);       // 8B-aligned (kb even, row 256B-aligned)
      d = wmma_f32(a, bfr[kk], d);
    }
#pragma unroll
    for (int r = 0; r < 8; ++r) {
      int o = mo * 16 + r + half * 8;
      yb[(size_t)o * (V_ * T_)] = d[r];
    }
  }
}

// ---------------------------------------------------------------------------
// Phase C: adjacency contraction as GEMM + fused BN + LeakyReLU.
// Per channel c: Z_c[v=25(pad32), m] = A^T[25 x 75(pad76)] x Y_c[75, m]
//   kw = k*25 + w, A^T[v,kw] = adjacency[kw*25 + v], Y_c[kw,m] = y[n,c*3+k,w,t].
// The adjacency A-fragments depend only on (lane, kstep, vtile): preload BOTH
// v-tiles' fragments into registers, then each of the 19 k-steps does one
// B-fragment load feeding two WMMAs -> y is read exactly once per element.
// ---------------------------------------------------------------------------
__global__ void MTA_adj_bn_kernel(const float* __restrict__ y,
                                  const float* __restrict__ Aadj,   // [3,25,25]
                                  const float* __restrict__ gamma,
                                  const float* __restrict__ beta,
                                  const float* __restrict__ mean,
                                  const float* __restrict__ var,
                                  float* __restrict__ outbuf, int br) {
  int lane = threadIdx.x & 31;
  int wg = blockIdx.x * (blockDim.x >> 5) + (threadIdx.x >> 5);
  // waves: c(64) x mt(512)
  int mt = wg % 512;
  int c  = wg / 512;

  int half = lane >> 4;
  int l15  = lane & 15;
  int col  = mt * 16 + l15;
  int n = col >> 7;
  int t = col & (T_ - 1);

  // ---- preload adjacency A-operand fragments for both v-tiles ----
  v2f afr[2][19];
#pragma unroll
  for (int vt = 0; vt < 2; ++vt) {
    int vrow = vt * 16 + l15;
#pragma unroll
    for (int kk = 0; kk < 19; ++kk) {
      int kb = kk * 4 + half * 2;
      afr[vt][kk].x = (kb     < 75 && vrow < V_) ? Aadj[(size_t)kb       * V_ + vrow] : 0.0f;
      afr[vt][kk].y = (kb + 1 < 75 && vrow < V_) ? Aadj[(size_t)(kb + 1) * V_ + vrow] : 0.0f;
    }
  }

  const float* Yc = y + ((size_t)n * OCH_ + c * 3) * (V_ * T_) + t;

  v8f d0 = {};
  v8f d1 = {};
#pragma unroll
  for (int kk = 0; kk < 19; ++kk) {
    int kb = kk * 4 + half * 2;
    bool ok0 = (kb     < 75);
    bool ok1 = (kb + 1 < 75);
    int k0 = kb / 25,       w0 = kb - k0 * 25;
    int k1 = (kb + 1) / 25, w1 = (kb + 1) - k1 * 25;
    v2f b;
    b.x = ok0 ? Yc[((size_t)k0 * V_ + w0) * T_] : 0.0f;
    b.y = ok1 ? Yc[((size_t)k1 * V_ + w1) * T_] : 0.0f;
    d0 = wmma_f32(afr[0][kk], b, d0);
    d1 = wmma_f32(afr[1][kk], b, d1);
  }

  float scale = gamma[br * C4_ + c] * rsqrtf(var[br * C4_ + c] + EPS_);
  float mu = mean[br * C4_ + c];
  float bt = beta[br * C4_ + c];
  float* ob = outbuf + (((size_t)n * C_ + br * C4_ + c) * V_) * T_ + t;

#pragma unroll
  for (int r = 0; r < 8; ++r) {
    int vv0 = r + half * 8;                   // tile 0 rows: always < 25
    float z0 = (d0[r] - mu) * scale + bt;
    z0 = z0 > 0.0f ? z0 : SLOPE_ * z0;
    ob[(size_t)vv0 * T_] = z0;

    int vv1 = 16 + r + half * 8;              // tile 1 rows: valid while < 25
    if (vv1 < V_) {
      float z1 = (d1[r] - mu) * scale + bt;
      z1 = z1 > 0.0f ? z1 : SLOPE_ * z1;
      ob[(size_t)vv1 * T_] = z1;
    }
  }
}

// ---------------------------------------------------------------------------
// Pass-through of split 3 (channels 192..255), 128-bit vectorized.
// ---------------------------------------------------------------------------
__global__ void MTA_copy_split3_kernel(const float* __restrict__ x,
                                       float* __restrict__ outbuf) {
  int idx = blockIdx.x * blockDim.x + threadIdx.x;   // over S_ELEMS/4 float4s
  if (idx >= S_ELEMS / 4) return;
  int n = idx / (C4_ * V_ * T_ / 4);
  int r = idx % (C4_ * V_ * T_ / 4);
  size_t off = (size_t)n * (C_ * V_ * T_ / 4) + 3 * (size_t)(C4_ * V_ * T_ / 4) + r;
  ((v4f*)outbuf)[off] = ((const v4f*)x)[off];
}

extern "C" void kernel_launch(void* const* d_in, const int* in_sizes, int n_in,
                              void* d_out, int out_size, void* d_ws, size_t ws_size,
                              hipStream_t stream) {
  const float* x     = (const float*)d_in[0];
  const float* Aadj  = (const float*)d_in[1];
  const float* cw    = (const float*)d_in[2];
  const float* cb    = (const float*)d_in[3];
  const float* gw    = (const float*)d_in[4];
  const float* gamma = (const float*)d_in[5];
  const float* beta  = (const float*)d_in[6];
  const float* mean  = (const float*)d_in[7];
  const float* var   = (const float*)d_in[8];
  float* out = (float*)d_out;

  float* s = (float*)d_ws;           // S_ELEMS floats
  float* y = s + S_ELEMS;            // Y_ELEMS floats

  MTA_copy_split3_kernel<<<S_ELEMS / 4 / 256, 256, 0, stream>>>(x, out);

  for (int br = 0; br < 3; ++br) {
    MTA_tconv_kernel<<<S_ELEMS / 256, 256, 0, stream>>>(x, out, cw, cb, s, br);
    // 25 * 512 = 12800 waves, 8 waves/block
    MTA_gemm_node_kernel<<<1600, 256, 0, stream>>>(s, gw, y, br);
    // 64 * 512 = 32768 waves, 8 waves/block
    MTA_adj_bn_kernel<<<4096, 256, 0, stream>>>(y, Aadj, gamma, beta, mean, var,
                                                out, br);
  }
}